// PanopticHead_1606317769399
// MI455X (gfx1250) — compile-verified
//
#include <hip/hip_runtime.h>
#include <hip/hip_bf16.h>
#include <stdint.h>

// Problem constants (from reference)
#define NBATCH 4
#define TBOX   32
#define CIN    133     // STUFF + THING
#define CSTUFF 53
#define COUT   85      // STUFF + T
#define HH     256
#define WW     512
#define MM     100     // mask resolution

typedef float    v4f __attribute__((ext_vector_type(4)));
typedef uint32_t v4u __attribute__((ext_vector_type(4)));
typedef int32_t  v4i __attribute__((ext_vector_type(4)));
typedef int32_t  v8i __attribute__((ext_vector_type(8)));

static __device__ __forceinline__ int imin(int a, int b) { return a < b ? a : b; }
static __device__ __forceinline__ int imax(int a, int b) { return a > b ? a : b; }

// ---------------------------------------------------------------------------
// Kernel A (emitted first so its asm is visible in the snippet):
// one block per (n,t) box. TDM-load the 100x100 f32 mask plane (40KB) into
// LDS, then each thread walks the box interior doing the bilinear paste
// (4 LDS gathers) + crop add, overwriting the zeros from the bulk kernel.
// ---------------------------------------------------------------------------
__global__ __launch_bounds__(256) void panoptic_paste(const float* __restrict__ sem,
                                                      const float* __restrict__ mask,
                                                      const float* __restrict__ boxes,
                                                      const int*   __restrict__ cls,
                                                      float*       __restrict__ out) {
    __shared__ float sm[MM * MM];

    const int nt = blockIdx.x;
    const int n  = nt / TBOX;
    const int t  = nt % TBOX;

    // --- TDM: DMA mask[n,t] (100x100 f32, fully contiguous) into LDS -------
    if (threadIdx.x < 32) {  // wave 0 issues; TENSORcnt is per-wave
        uint64_t ga  = (uint64_t)(uintptr_t)(mask + (size_t)nt * (MM * MM));
        uint32_t lds = (uint32_t)(uintptr_t)(void*)sm;  // flat LDS addr low 32 bits
        // D# group 0: [1:0]=count=1 | lds_addr | global_addr[56:0] | type=2
        v4u g0 = { 1u,
                   lds,
                   (uint32_t)ga,
                   ((uint32_t)(ga >> 32) & 0x01FFFFFFu) | 0x80000000u };
        // D# group 1: data_size=4B; tensor_dim0/1=100; tile_dim0/1=100;
        //             tensor_dim0_stride=100; no multicast/pad/iterate.
        v8i g1 = { (int)(2u << 16),        // [17:16] data_size = 2 (4 bytes)
                   (int)(MM << 16),        // [63:48]  tensor_dim0 lo16
                   (int)(MM << 16),        // [95:80]  tensor_dim1 lo16 (dim0 hi16 = 0)
                   (int)(MM << 16),        // [127:112] tile_dim0   (dim1 hi16 = 0)
                   MM,                     // [143:128] tile_dim1
                   MM,                     // [191:160] tensor_dim0_stride lo32
                   0, 0 };
        v4i gz4 = { 0, 0, 0, 0 };          // groups 2/3: 2D tile, all unused
        v8i gz8 = { 0, 0, 0, 0, 0, 0, 0, 0 };
        __builtin_amdgcn_tensor_load_to_lds(g0, g1, gz4, gz4, gz8, 0);
        __builtin_amdgcn_s_wait_tensorcnt(0);
    }
    __syncthreads();

    // --- Box geometry (matches reference int semantics exactly) ------------
    const float bx1 = boxes[nt * 4 + 0], by1 = boxes[nt * 4 + 1];
    const float bx2 = boxes[nt * 4 + 2], by2 = boxes[nt * 4 + 3];
    const int x0 = (int)bx1, y0 = (int)by1;          // truncation == astype(int32), boxes >= 0
    const int x2 = (int)bx2, y2 = (int)by2;
    const int xr = (int)rintf(bx2) + 1;              // rintf == jnp.round (half-to-even)
    const int yr = (int)rintf(by2) + 1;
    const float sx = 100.0f / (float)(x2 - x0 + 1);
    const float sy = 100.0f / (float)(y2 - y0 + 1);

    const int lox   = imax(x0, 0),      loy   = imax(y0, 0);
    const int px_hi = imin(x2 + 1, WW), py_hi = imin(y2 + 1, HH);  // paste window
    const int cx_hi = imin(xr, WW),     cy_hi = imin(yr, HH);      // crop window
    const int hix   = imax(px_hi, cx_hi), hiy = imax(py_hi, cy_hi);
    const int rw = hix - lox, rh = hiy - loy;
    if (rw <= 0 || rh <= 0) return;

    const int c = cls[nt];
    const float* semc = sem + (size_t)(n * CIN + CSTUFF + c) * (HH * WW);
    float*       outc = out + (size_t)(n * COUT + CSTUFF + t) * (HH * WW);

    const int total = rw * rh;
    for (int i = threadIdx.x; i < total; i += 256) {
        const int yy = loy + i / rw;
        const int xx = lox + i % rw;
        float v = 0.0f;
        if (xx < px_hi && yy < py_hi) {
            float srcx = ((float)(xx - x0) + 0.5f) * sx - 0.5f;
            srcx = fminf(fmaxf(srcx, 0.0f), 99.0f);
            const int   ix0 = (int)srcx;             // srcx >= 0 -> trunc == floor
            const float fx  = srcx - (float)ix0;
            const int   ix1 = imin(ix0 + 1, MM - 1);
            float srcy = ((float)(yy - y0) + 0.5f) * sy - 0.5f;
            srcy = fminf(fmaxf(srcy, 0.0f), 99.0f);
            const int   iy0 = (int)srcy;
            const float fy  = srcy - (float)iy0;
            const int   iy1 = imin(iy0 + 1, MM - 1);
            const float* r0 = sm + iy0 * MM;
            const float* r1 = sm + iy1 * MM;
            const float a = (1.0f - fx) * r0[ix0] + fx * r0[ix1];
            const float b = (1.0f - fx) * r1[ix0] + fx * r1[ix1];
            v = (1.0f - fy) * a + fy * b;
        }
        if (xx < cx_hi && yy < cy_hi) v += semc[yy * WW + xx];
        outc[yy * WW + xx] = v;
    }
}

// ---------------------------------------------------------------------------
// Kernel B: streaming bulk pass. Copies the 53 stuff channels and zero-fills
// the 32 thing channels. float4 (b128) accesses, non-temporal (NT) hints so
// this one-touch 290MB stream does not pollute L2 for the paste kernel.
// ---------------------------------------------------------------------------
__global__ __launch_bounds__(256) void panoptic_bulk(const v4f* __restrict__ sem,
                                                     v4f* __restrict__ out) {
    constexpr int P4  = (HH * WW) / 4;        // 32768 float4 per plane
    constexpr int TOT = NBATCH * COUT * P4;   // 11,141,120
    int gid = blockIdx.x * 256 + threadIdx.x;
    if (gid >= TOT) return;
    int p  = gid & (P4 - 1);
    int nc = gid >> 15;            // n*COUT + c
    int c  = nc % COUT;
    int n  = nc / COUT;
    v4f v;
    if (c < CSTUFF) {
        size_t src = ((size_t)(n * CIN + c) << 15) + (size_t)p;
        v = __builtin_nontemporal_load(&sem[src]);
    } else {
        v = (v4f){0.0f, 0.0f, 0.0f, 0.0f};
    }
    __builtin_nontemporal_store(v, &out[gid]);
}

// ---------------------------------------------------------------------------
extern "C" void kernel_launch(void* const* d_in, const int* in_sizes, int n_in,
                              void* d_out, int out_size, void* d_ws, size_t ws_size,
                              hipStream_t stream) {
    const float* sem   = (const float*)d_in[0];  // (4,133,256,512) f32
    const float* mask  = (const float*)d_in[1];  // (4,32,1,100,100) f32
    const float* boxes = (const float*)d_in[2];  // (4,32,4) f32
    const int*   cls   = (const int*)d_in[3];    // (4,32) i32
    float* out = (float*)d_out;                  // (4,85,256,512) f32

    constexpr int TOT4 = NBATCH * COUT * (HH * WW) / 4;  // 11,141,120
    hipLaunchKernelGGL(panoptic_bulk, dim3((TOT4 + 255) / 256), dim3(256), 0, stream,
                       (const v4f*)sem, (v4f*)out);
    hipLaunchKernelGGL(panoptic_paste, dim3(NBATCH * TBOX), dim3(256), 0, stream,
                       sem, mask, boxes, cls, out);
}